// TimedGINConv_15444702396461
// MI455X (gfx1250) — compile-verified
//
#include <hip/hip_runtime.h>
#include <hip/hip_bf16.h>

typedef __attribute__((ext_vector_type(2))) float v2f;
typedef __attribute__((ext_vector_type(8))) float v8f;

#define FDIM 256

// ---------------------------------------------------------------------------
// Kernel 1: h = (1 + eps) * feat   (eps = 0.1), vectorized float4
// ---------------------------------------------------------------------------
__global__ void gin_init_kernel(const float4* __restrict__ feat,
                                float4* __restrict__ h, int n4) {
    int i = blockIdx.x * blockDim.x + threadIdx.x;
    if (i < n4) {
        float4 f = feat[i];
        float4 r;
        r.x = 1.1f * f.x;
        r.y = 1.1f * f.y;
        r.z = 1.1f * f.z;
        r.w = 1.1f * f.w;
        h[i] = r;
    }
}

// ---------------------------------------------------------------------------
// Kernel 2: scatter-add edge messages: h[dst] += feat[src]
// One block per edge, one thread per channel -> one f32 atomic per thread.
// feat (10 MB) is L2-resident on MI455X (192 MB L2), so gathers are L2 hits
// and the atomics resolve at the L2 atomic units.
// ---------------------------------------------------------------------------
__global__ void gin_scatter_kernel(const float* __restrict__ feat,
                                   const int* __restrict__ src,
                                   const int* __restrict__ dst,
                                   float* __restrict__ h) {
    const int e = blockIdx.x;
    const int c = threadIdx.x;
    const int s = src[e];   // uniform per block -> scalar load
    const int d = dst[e];
    atomicAdd(h + (size_t)d * FDIM + c, feat[(size_t)s * FDIM + c]);
}

// ---------------------------------------------------------------------------
// Kernel 3: out = h @ W using V_WMMA_F32_16X16X4_F32 (fp32-exact path).
// Block = 256 threads = 8 wave32 waves. Block owns 16 output rows; wave w
// owns columns [32w, 32w+32) as two 16x16 fp32 accumulator tiles.
// K loop: 64 steps of K=4, 2 WMMAs/step.
//
// A layout (ISA 7.12.2, 32-bit A 16x4): lane = K[1]*16 + M, vgpr = K[0]
//   -> lane (half = lane>>4, m = lane&15) loads contiguous float2 at
//      h[(row0+m)*256 + k0 + 2*half].
// B layout (4x16, row striped across lanes): vgpr v, half hh holds
//   B[k0 + 2*hh + v][n0 + (lane&15)].
// C/D layout: vgpr v -> row (v + 8*half), col n0 + (lane&15).
// ---------------------------------------------------------------------------
__global__ void gin_gemm_wmma_kernel(const float* __restrict__ h,
                                     const float* __restrict__ w,
                                     float* __restrict__ out) {
    const int lane = threadIdx.x & 31;
    const int wave = threadIdx.x >> 5;          // 0..7
    const int row0 = blockIdx.x * 16;
    const int n0   = wave * 32;                 // 2 column tiles per wave
    const int half = lane >> 4;                 // 0 or 1
    const int m    = lane & 15;                 // also the N index for B/C/D

    v8f c0 = {};
    v8f c1 = {};

    const float* arow = h + (size_t)(row0 + m) * FDIM;

#pragma unroll 4
    for (int k0 = 0; k0 < FDIM; k0 += 4) {
        const int ka = k0 + 2 * half;

        v2f a;
        a.x = arow[ka + 0];
        a.y = arow[ka + 1];

        const float* wr0 = w + (size_t)ka * FDIM + n0 + m;
        const float* wr1 = wr0 + FDIM;

        v2f b0, b1;
        b0.x = wr0[0];
        b0.y = wr1[0];
        b1.x = wr0[16];
        b1.y = wr1[16];

        c0 = __builtin_amdgcn_wmma_f32_16x16x4_f32(
                 /*neg_a=*/false, a, /*neg_b=*/false, b0,
                 /*c_mod=*/(short)0, c0, /*reuse_a=*/false, /*reuse_b=*/false);
        c1 = __builtin_amdgcn_wmma_f32_16x16x4_f32(
                 /*neg_a=*/false, a, /*neg_b=*/false, b1,
                 /*c_mod=*/(short)0, c1, /*reuse_a=*/false, /*reuse_b=*/false);
    }

    float* obase = out + (size_t)row0 * FDIM + n0 + m;
#pragma unroll
    for (int v = 0; v < 8; ++v) {
        const int mr = v + 8 * half;
        obase[(size_t)mr * FDIM + 0]  = c0[v];
        obase[(size_t)mr * FDIM + 16] = c1[v];
    }
}

// ---------------------------------------------------------------------------
extern "C" void kernel_launch(void* const* d_in, const int* in_sizes, int n_in,
                              void* d_out, int out_size, void* d_ws, size_t ws_size,
                              hipStream_t stream) {
    const float* feat   = (const float*)d_in[0];
    const int*   src    = (const int*)d_in[1];
    const int*   dst    = (const int*)d_in[2];
    const float* weight = (const float*)d_in[3];
    float*       out    = (float*)d_out;
    float*       hbuf   = (float*)d_ws;        // [M, 256] fp32 staging

    const int nfeat  = in_sizes[0];            // M * 256
    const int nedges = in_sizes[1];
    const int M      = nfeat / FDIM;

    // 1) h = 1.1 * feat
    const int n4 = nfeat / 4;
    gin_init_kernel<<<(n4 + 255) / 256, 256, 0, stream>>>(
        (const float4*)feat, (float4*)hbuf, n4);

    // 2) h[dst] += feat[src] over all edges
    gin_scatter_kernel<<<nedges, FDIM, 0, stream>>>(feat, src, dst, hbuf);

    // 3) out = h @ W  (fp32 WMMA), 16 rows per block, exact tiling (M % 16 == 0)
    gin_gemm_wmma_kernel<<<M / 16, 256, 0, stream>>>(hbuf, weight, out);
}